// SparseWavefunction_31911607009438
// MI455X (gfx1250) — compile-verified
//
#include <hip/hip_runtime.h>
#include <hip/hip_bf16.h>

// ---------------------------------------------------------------------------
// CDNA5 (gfx1250) implementation of the sparse-wavefunction forward pass.
// All dense 64-K GEMMs run on the matrix pipe via V_WMMA_F32_16X16X4_F32
// (exact fp32, matching the fp32 reference), chained 16x along K=64.
// ---------------------------------------------------------------------------

typedef __attribute__((ext_vector_type(2))) float v2f;
typedef __attribute__((ext_vector_type(8))) float v8f;

static constexpr int NELEC = 4096;
static constexpr int NK    = 64;    // neighbours
static constexpr int WID   = 64;    // width
static constexpr int NORB  = 4096;
static constexpr int NBH   = 16;
static constexpr int NBO   = 8;
static constexpr int NBE   = 16;
static constexpr int LDSS  = 66;    // padded LDS row stride (conflict-free)

__device__ __forceinline__ v8f wmma4(v2f a, v2f b, v8f c) {
  // D = A(16x4,f32) * B(4x16,f32) + C(16x16,f32)  -- exact fp32
  return __builtin_amdgcn_wmma_f32_16x16x4_f32(
      /*neg_a=*/false, a, /*neg_b=*/false, b,
      /*c_mod=*/(short)0, c, /*reuse_a=*/false, /*reuse_b=*/false);
}

__device__ __forceinline__ float silu_f(float x) {
  return x / (1.0f + __expf(-x));
}

// ---------------------------------------------------------------------------
// Kernel A: per-electron pairwise features.
//   beta[n,k,8]  (stored to ws for the message-passing layer)
//   gin = silu([diff,dist] @ g_W1 + b1)              (LDS, 64x64)
//   g2  = silu(gin @ g_W2 + b2)                      (WMMA, 64x64x64)
//   fk0 = beta @ Gamma0                              (LDS, 64x64)
//   h0[n,w] = sum_k g2[k,w] * fk0[k,w]               (tile column-reduce)
// One block (256 thr = 8 waves) per electron n.
// ---------------------------------------------------------------------------
__global__ __launch_bounds__(256) void ka_pairwise(
    const float* __restrict__ r, const float* __restrict__ wgt,
    const float* __restrict__ scales,
    const float* __restrict__ env_W, const float* __restrict__ env_b,
    const float* __restrict__ pw_W1, const float* __restrict__ pw_b1,
    const float* __restrict__ pw_W2, const float* __restrict__ pw_b2,
    const float* __restrict__ g_W1, const float* __restrict__ g_b1,
    const float* __restrict__ g_W2, const float* __restrict__ g_b2,
    const float* __restrict__ Gamma0, const int* __restrict__ ind,
    float* __restrict__ h0_out, float* __restrict__ beta_out)
{
  __shared__ float s_diff[NK][4];        // dx,dy,dz,dist
  __shared__ float s_beta[NK][NBO];
  __shared__ float s_gin[NK][LDSS];
  __shared__ float s_fk0[NK][LDSS];
  __shared__ float s_h0[WID];

  const int n   = blockIdx.x;
  const int tid = threadIdx.x;

  if (tid < WID) s_h0[tid] = 0.0f;

  const float rx = r[n * 3 + 0], ry = r[n * 3 + 1], rz = r[n * 3 + 2];

  if (tid < NK) {
    const int k  = tid;
    const int nb = ind[n * NK + k];
    const float dx = rx - r[nb * 3 + 0];
    const float dy = ry - r[nb * 3 + 1];
    const float dz = rz - r[nb * 3 + 2];
    const float dist = sqrtf(dx * dx + dy * dy + dz * dz);
    s_diff[k][0] = dx; s_diff[k][1] = dy; s_diff[k][2] = dz; s_diff[k][3] = dist;

    // envelope: exp(-d^2/softplus(scale)) @ env_W + env_b
    float env[NBO];
    #pragma unroll
    for (int j = 0; j < NBO; ++j) env[j] = env_b[j];
    #pragma unroll
    for (int e = 0; e < NBE; ++e) {
      const float s  = scales[e];
      const float sp = (s > 20.0f) ? s : log1pf(__expf(s));
      const float ev = __expf(-dist * dist / sp);
      #pragma unroll
      for (int j = 0; j < NBO; ++j) env[j] += ev * env_W[e * NBO + j];
    }
    // pairwise MLP: silu(diff @ pw_W1 + b1) @ pw_W2 + b2
    float hmid[NBH];
    #pragma unroll
    for (int m = 0; m < NBH; ++m) {
      const float a = pw_b1[m] + dx * pw_W1[0 * NBH + m]
                               + dy * pw_W1[1 * NBH + m]
                               + dz * pw_W1[2 * NBH + m];
      hmid[m] = silu_f(a);
    }
    float dvec[NBO];
    #pragma unroll
    for (int j = 0; j < NBO; ++j) {
      float a = pw_b2[j];
      #pragma unroll
      for (int m = 0; m < NBH; ++m) a += hmid[m] * pw_W2[m * NBO + j];
      dvec[j] = a;
    }
    // polynomial cutoff (p=4): 1 - 15 d^4 + 24 d^5 - 10 d^6, d = dist/5
    const float dd = dist * 0.2f;
    float cut = 0.0f;
    if (dd < 1.0f) {
      const float d2 = dd * dd;
      const float d4 = d2 * d2;
      cut = 1.0f - 15.0f * d4 + 24.0f * d4 * dd - 10.0f * d4 * d2;
    }
    const float wg = wgt[n * NK + k];
    #pragma unroll
    for (int j = 0; j < NBO; ++j) {
      const float b = env[j] * dvec[j] * cut * wg;
      s_beta[k][j] = b;
      beta_out[(n * NK + k) * NBO + j] = b;
    }
  }
  __syncthreads();

  // gin (silu of 4->64 layer) and fk0 = beta @ Gamma0, 16 columns per thread
  {
    const int k  = tid >> 2;
    const int w0 = (tid & 3) * 16;
    const float dx = s_diff[k][0], dy = s_diff[k][1];
    const float dz = s_diff[k][2], dd = s_diff[k][3];
    #pragma unroll
    for (int w = w0; w < w0 + 16; ++w) {
      const float a = g_b1[w] + dx * g_W1[0 * WID + w] + dy * g_W1[1 * WID + w]
                              + dz * g_W1[2 * WID + w] + dd * g_W1[3 * WID + w];
      s_gin[k][w] = silu_f(a);
    }
    #pragma unroll
    for (int w = w0; w < w0 + 16; ++w) {
      float a = 0.0f;
      #pragma unroll
      for (int j = 0; j < NBO; ++j) a += s_beta[k][j] * Gamma0[j * WID + w];
      s_fk0[k][w] = a;
    }
  }
  __syncthreads();

  // WMMA: g2 = silu(gin @ g_W2 + b2), then h0 += column-sum(g2 * fk0)
  {
    const int wave = tid >> 5;
    const int lane = tid & 31;
    const int half = lane >> 4;
    const int idx  = lane & 15;
    for (int t = wave * 2; t < wave * 2 + 2; ++t) {
      const int pt = t >> 2;     // pair (row) tile
      const int nt = t & 3;      // width (col) tile
      v8f acc = {};
      #pragma unroll
      for (int k0 = 0; k0 < NK; k0 += 4) {
        v2f a, b;
        a.x = s_gin[pt * 16 + idx][k0 + 2 * half];
        a.y = s_gin[pt * 16 + idx][k0 + 2 * half + 1];
        b.x = g_W2[(k0 + 2 * half) * WID + nt * 16 + idx];
        b.y = g_W2[(k0 + 2 * half + 1) * WID + nt * 16 + idx];
        acc = wmma4(a, b, acc);
      }
      const int col  = nt * 16 + idx;
      const float bias = g_b2[col];
      float csum = 0.0f;
      #pragma unroll
      for (int i = 0; i < 8; ++i) {
        const int row = pt * 16 + i + 8 * half;
        const float v = silu_f(acc[i] + bias);
        csum += v * s_fk0[row][col];
      }
      csum += __shfl_xor(csum, 16, 32);   // combine half-waves (full 16 rows)
      if (half == 0) atomicAdd(&s_h0[col], csum);  // ds_add_f32
    }
  }
  __syncthreads();
  if (tid < WID) h0_out[n * WID + tid] = s_h0[tid];
}

// ---------------------------------------------------------------------------
// Kernel B: main MLP, 3 fused 64x64 layers (row-local), LDS ping-pong.
// One block per 64 rows; 8 waves x 2 tiles/layer.
// ---------------------------------------------------------------------------
__global__ __launch_bounds__(256) void kb_mlp(
    const float* __restrict__ h0,
    const float* __restrict__ W0, const float* __restrict__ W1,
    const float* __restrict__ W2,
    const float* __restrict__ b0, const float* __restrict__ b1,
    const float* __restrict__ b2,
    float* __restrict__ h_out)
{
  __shared__ float bufA[64][LDSS];
  __shared__ float bufB[64][LDSS];
  const int tid  = threadIdx.x;
  const int row0 = blockIdx.x * 64;

  for (int i = tid; i < 64 * WID; i += 256)
    bufA[i >> 6][i & 63] = h0[row0 * WID + i];
  __syncthreads();

  const float* Ws[3] = {W0, W1, W2};
  const float* bs[3] = {b0, b1, b2};
  const int wave = tid >> 5, lane = tid & 31;
  const int half = lane >> 4, idx = lane & 15;

  float (*src)[LDSS] = bufA;
  float (*dst)[LDSS] = bufB;
  for (int l = 0; l < 3; ++l) {
    const float* Wm = Ws[l];
    const float* bv = bs[l];
    for (int t = wave * 2; t < wave * 2 + 2; ++t) {
      const int pt = t >> 2, nt = t & 3;
      v8f acc = {};
      #pragma unroll
      for (int k0 = 0; k0 < WID; k0 += 4) {
        v2f a, b;
        a.x = src[pt * 16 + idx][k0 + 2 * half];
        a.y = src[pt * 16 + idx][k0 + 2 * half + 1];
        b.x = Wm[(k0 + 2 * half) * WID + nt * 16 + idx];
        b.y = Wm[(k0 + 2 * half + 1) * WID + nt * 16 + idx];
        acc = wmma4(a, b, acc);
      }
      const int col  = nt * 16 + idx;
      const float bias = bv[col];
      #pragma unroll
      for (int i = 0; i < 8; ++i) {
        const int row = pt * 16 + i + 8 * half;
        float v = acc[i] + bias;
        if (l < 2) v = silu_f(v);
        dst[row][col] = v;
      }
    }
    __syncthreads();
    float (*tmp)[LDSS] = src; src = dst; dst = tmp;
  }
  for (int i = tid; i < 64 * WID; i += 256)
    h_out[row0 * WID + i] = src[i >> 6][i & 63];
}

// ---------------------------------------------------------------------------
// Kernel C: message passing. msg[n,w] = sum_k (beta@Gamma1)[n,k,w] * h[ind[n,k],w]
// Gather-bound: stage neighbour rows + fk1^T in LDS, then reduce over k.
// One block per electron n.
// ---------------------------------------------------------------------------
__global__ __launch_bounds__(256) void kc_msg(
    const float* __restrict__ h, const float* __restrict__ beta,
    const float* __restrict__ Gamma1, const int* __restrict__ ind,
    float* __restrict__ msg_out)
{
  __shared__ int   s_ind[NK];
  __shared__ float s_hnb[NK][LDSS];   // [k][w]
  __shared__ float s_fk1t[WID][LDSS]; // [w][k] (transposed for the reduction)
  __shared__ float s_part[4][WID];

  const int n = blockIdx.x, tid = threadIdx.x;
  if (tid < NK) s_ind[tid] = ind[n * NK + tid];
  __syncthreads();
  {
    const int k  = tid >> 2;
    const int c0 = (tid & 3) * 16;
    const int nb = s_ind[k];
    for (int c = c0; c < c0 + 16; ++c) s_hnb[k][c] = h[nb * WID + c];
    float bb[NBO];
    #pragma unroll
    for (int j = 0; j < NBO; ++j) bb[j] = beta[(n * NK + k) * NBO + j];
    for (int w = c0; w < c0 + 16; ++w) {
      float a = 0.0f;
      #pragma unroll
      for (int j = 0; j < NBO; ++j) a += bb[j] * Gamma1[j * WID + w];
      s_fk1t[w][k] = a;
    }
  }
  __syncthreads();
  {
    const int w = tid & 63, q = tid >> 6;
    float p = 0.0f;
    #pragma unroll
    for (int k = q * 16; k < q * 16 + 16; ++k)
      p += s_fk1t[w][k] * s_hnb[k][w];
    s_part[q][w] = p;
  }
  __syncthreads();
  if (tid < WID)
    msg_out[n * WID + tid] =
        s_part[0][tid] + s_part[1][tid] + s_part[2][tid] + s_part[3][tid];
}

// ---------------------------------------------------------------------------
// Kernel D: h_out = silu(h0 @ proj_W + proj_b + msg). One 64x64x64 WMMA GEMM.
// ---------------------------------------------------------------------------
__global__ __launch_bounds__(256) void kd_hout(
    const float* __restrict__ h0, const float* __restrict__ projW,
    const float* __restrict__ projb, const float* __restrict__ msg,
    float* __restrict__ hout)
{
  __shared__ float bufA[64][LDSS];
  __shared__ float bufB[64][LDSS];
  const int tid = threadIdx.x;
  const int row0 = blockIdx.x * 64;
  for (int i = tid; i < 64 * WID; i += 256)
    bufA[i >> 6][i & 63] = h0[row0 * WID + i];
  __syncthreads();
  const int wave = tid >> 5, lane = tid & 31;
  const int half = lane >> 4, idx = lane & 15;
  for (int t = wave * 2; t < wave * 2 + 2; ++t) {
    const int pt = t >> 2, nt = t & 3;
    v8f acc = {};
    #pragma unroll
    for (int k0 = 0; k0 < WID; k0 += 4) {
      v2f a, b;
      a.x = bufA[pt * 16 + idx][k0 + 2 * half];
      a.y = bufA[pt * 16 + idx][k0 + 2 * half + 1];
      b.x = projW[(k0 + 2 * half) * WID + nt * 16 + idx];
      b.y = projW[(k0 + 2 * half + 1) * WID + nt * 16 + idx];
      acc = wmma4(a, b, acc);
    }
    const int col = nt * 16 + idx;
    const float bias = projb[col];
    #pragma unroll
    for (int i = 0; i < 8; ++i) {
      const int row = pt * 16 + i + 8 * half;
      const float v = acc[i] + bias + msg[(row0 + row) * WID + col];
      bufB[row][col] = silu_f(v);
    }
  }
  __syncthreads();
  for (int i = tid; i < 64 * WID; i += 256)
    hout[row0 * WID + i] = bufB[i >> 6][i & 63];
}

// ---------------------------------------------------------------------------
// Kernel E: phi = (h_out @ orb_W + orb_b) * exp(-0.2 * |r - R_orb|)
// 4096x4096x64 WMMA GEMM fused with the orbital envelope + 64MB store.
// Block = 8 waves covering one 16x128 output strip; A tile shared via LDS.
// ---------------------------------------------------------------------------
__global__ __launch_bounds__(256) void ke_orb(
    const float* __restrict__ hout, const float* __restrict__ orbW,
    const float* __restrict__ orbb, const float* __restrict__ r,
    const float* __restrict__ R_orb, float* __restrict__ phi)
{
  __shared__ float sA[16][LDSS];
  __shared__ float sr[16][3];
  const int tid = threadIdx.x;
  const int bid = blockIdx.x;
  const int mt = bid & 255;   // 256 M tiles of 16 rows
  const int ng = bid >> 8;    // 32 N groups of 128 cols
  const int m0 = mt * 16;

  for (int i = tid; i < 16 * WID; i += 256)
    sA[i >> 6][i & 63] = hout[m0 * WID + i];
  if (tid < 48) sr[tid / 3][tid % 3] = r[m0 * 3 + tid];
  __syncthreads();

  const int wave = tid >> 5, lane = tid & 31;
  const int half = lane >> 4, idx = lane & 15;
  const int n0 = ng * 128 + wave * 16;

  v8f acc = {};
  #pragma unroll
  for (int k0 = 0; k0 < WID; k0 += 4) {
    v2f a, b;
    a.x = sA[idx][k0 + 2 * half];
    a.y = sA[idx][k0 + 2 * half + 1];
    b.x = orbW[(k0 + 2 * half) * NORB + n0 + idx];
    b.y = orbW[(k0 + 2 * half + 1) * NORB + n0 + idx];
    acc = wmma4(a, b, acc);
  }
  const int col = n0 + idx;
  const float bias = orbb[col];
  const float Rx = R_orb[col * 3 + 0];
  const float Ry = R_orb[col * 3 + 1];
  const float Rz = R_orb[col * 3 + 2];
  #pragma unroll
  for (int i = 0; i < 8; ++i) {
    const int row = i + 8 * half;
    const float dx = sr[row][0] - Rx;
    const float dy = sr[row][1] - Ry;
    const float dz = sr[row][2] - Rz;
    const float dist = sqrtf(dx * dx + dy * dy + dz * dz);
    phi[(m0 + row) * NORB + col] = (acc[i] + bias) * __expf(-0.2f * dist);
  }
}

// ---------------------------------------------------------------------------
// Host launcher
// ---------------------------------------------------------------------------
extern "C" void kernel_launch(void* const* d_in, const int* in_sizes, int n_in,
                              void* d_out, int out_size, void* d_ws, size_t ws_size,
                              hipStream_t stream) {
  // setup_inputs() flattening order:
  const float* r       = (const float*)d_in[0];   // [N,3]
  const float* wgt     = (const float*)d_in[1];   // [N,K]
  const float* R_orb   = (const float*)d_in[2];   // [NORB,3]
  const float* scales  = (const float*)d_in[3];   // [BE]
  const float* env_W   = (const float*)d_in[4];   // [BE,BO]
  const float* env_b   = (const float*)d_in[5];   // [BO]
  const float* pw_W1   = (const float*)d_in[6];   // [3,BH]
  const float* pw_b1   = (const float*)d_in[7];   // [BH]
  const float* pw_W2   = (const float*)d_in[8];   // [BH,BO]
  const float* pw_b2   = (const float*)d_in[9];   // [BO]
  const float* g_W1    = (const float*)d_in[10];  // [4,W]
  const float* g_b1    = (const float*)d_in[11];  // [W]
  const float* g_W2    = (const float*)d_in[12];  // [W,W]
  const float* g_b2    = (const float*)d_in[13];  // [W]
  const float* Gamma0  = (const float*)d_in[14];  // [BO,W]
  const float* mlp_W0  = (const float*)d_in[15];  // [W,W]
  const float* mlp_W1  = (const float*)d_in[16];
  const float* mlp_W2  = (const float*)d_in[17];
  const float* mlp_b0  = (const float*)d_in[18];  // [W]
  const float* mlp_b1  = (const float*)d_in[19];
  const float* mlp_b2  = (const float*)d_in[20];
  const float* proj_W  = (const float*)d_in[21];  // [W,W]
  const float* proj_b  = (const float*)d_in[22];  // [W]
  const float* Gamma1  = (const float*)d_in[23];  // [BO,W]
  const float* orb_W   = (const float*)d_in[24];  // [W,NORB]
  const float* orb_b   = (const float*)d_in[25];  // [NORB]
  const int*   ind     = (const int*)  d_in[26];  // [N,K]

  float* phi = (float*)d_out;                     // [N, NORB]

  // workspace layout (floats)
  float* ws    = (float*)d_ws;
  float* h0    = ws;                              // N*W      = 262144
  float* h     = ws + 262144;                     // N*W
  float* msg   = ws + 524288;                     // N*W
  float* h_out = ws + 786432;                     // N*W
  float* beta  = ws + 1048576;                    // N*K*BO   = 2097152

  ka_pairwise<<<NELEC, 256, 0, stream>>>(
      r, wgt, scales, env_W, env_b, pw_W1, pw_b1, pw_W2, pw_b2,
      g_W1, g_b1, g_W2, g_b2, Gamma0, ind, h0, beta);

  kb_mlp<<<NELEC / 64, 256, 0, stream>>>(
      h0, mlp_W0, mlp_W1, mlp_W2, mlp_b0, mlp_b1, mlp_b2, h);

  kc_msg<<<NELEC, 256, 0, stream>>>(h, beta, Gamma1, ind, msg);

  kd_hout<<<NELEC / 64, 256, 0, stream>>>(h0, proj_W, proj_b, msg, h_out);

  ke_orb<<<(NELEC / 16) * (NORB / 128), 256, 0, stream>>>(
      h_out, orb_W, orb_b, r, R_orb, phi);
}